// HungarianMatcher_8418135900637
// MI455X (gfx1250) — compile-verified
//
#include <hip/hip_runtime.h>
#include <math.h>
#include <stdint.h>

#define Bn 16
#define Qn 1024
#define Tn 128
#define Cn 1024
#define INFF 1e9f

typedef float v2f  __attribute__((ext_vector_type(2)));
typedef float v8f  __attribute__((ext_vector_type(8)));

// ---------------------------------------------------------------------------
// Kernel A: per-(b,q) softmax stats + probability gathered at the T label
// positions. One wave32 per query row. gp[b][q][t] = softmax(logits[b,q])[lab[t]]
// ---------------------------------------------------------------------------
__global__ void __launch_bounds__(256) gather_prob_kernel(
    const float* __restrict__ logits,   // [B,Q,C]
    const int*   __restrict__ labels,   // [B,T]
    float*       __restrict__ gp)       // [B,Q,T]
{
  const int wave = (blockIdx.x << 3) + (threadIdx.x >> 5);
  const int lane = threadIdx.x & 31;
  const int b = wave / Qn;
  const int q = wave % Qn;
  const float* row = logits + (size_t)(b * Qn + q) * Cn;

  float vals[32];
  float m = -INFF;
#pragma unroll
  for (int k = 0; k < 32; ++k) {
    vals[k] = row[lane + (k << 5)];
    m = fmaxf(m, vals[k]);
  }
#pragma unroll
  for (int off = 16; off > 0; off >>= 1) m = fmaxf(m, __shfl_xor(m, off, 32));

  float s = 0.0f;
#pragma unroll
  for (int k = 0; k < 32; ++k) s += __expf(vals[k] - m);
#pragma unroll
  for (int off = 16; off > 0; off >>= 1) s += __shfl_xor(s, off, 32);
  const float inv_s = 1.0f / s;

  float* out = gp + (size_t)(b * Qn + q) * Tn;
  const int* lab = labels + b * Tn;
#pragma unroll
  for (int t = lane; t < Tn; t += 32) {
    const int c = lab[t];
    out[t] = __expf(row[c] - m) * inv_s;
  }
}

// ---------------------------------------------------------------------------
// Kernel B: cost via WMMA. One wave per 16-query strip; the A fragment (16x4
// coords) and the 8 per-lane |p|^2 values are loaded once and reused across
// all 8 t-tiles -> 8 back-to-back V_WMMA_F32_16X16X4_F32 per wave.
// cost = sqrt(|p|^2 + |t|^2 - 2 p.t) - prob. Writes C[b][q][t] and CT[b][t][q].
// ---------------------------------------------------------------------------
__global__ void __launch_bounds__(256) cost_wmma_kernel(
    const float* __restrict__ pc,   // [B,Q,4]
    const float* __restrict__ tc,   // [B,T,4]
    const float* __restrict__ gp,   // [B,Q,T]
    float*       __restrict__ Cout, // [B,Q,T]
    float*       __restrict__ CT)   // [B,T,Q]
{
  const int wave = (blockIdx.x << 3) + (threadIdx.x >> 5);  // 0 .. B*64-1
  const int lane = threadIdx.x & 31;
  const int mt = wave & 63;         // q-tile (Q/16 = 64)
  const int b  = wave >> 6;
  const int q0 = mt << 4;
  const int half = lane >> 4;       // 16-lane half
  const int mr   = lane & 15;       // M row for A, N col for B/D
  const int kb   = half << 1;       // K pair held by this half

  // A fragment: lanes 0-15 hold M=mr K={0,1}; lanes 16-31 hold M=mr K={2,3}
  const float* pa = pc + (size_t)(b * Qn + q0 + mr) * 4 + kb;
  v2f a;
  a.x = pa[0];
  a.y = pa[1];

  // |p|^2 for the 8 output rows this lane owns (D: M = r + half*8, N = mr)
  float pn[8];
#pragma unroll
  for (int r = 0; r < 8; ++r) {
    const int q = q0 + r + (half << 3);
    const float4 pcv = *(const float4*)(pc + (size_t)(b * Qn + q) * 4);
    pn[r] = pcv.x * pcv.x + pcv.y * pcv.y + pcv.z * pcv.z + pcv.w * pcv.w;
  }

#pragma unroll
  for (int nt = 0; nt < 8; ++nt) {
    const int t0 = nt << 4;
    const float* pb = tc + (size_t)(b * Tn + t0 + mr) * 4 + kb;
    v2f bm;
    bm.x = pb[0];
    bm.y = pb[1];
    v8f acc = {};
    v8f d = __builtin_amdgcn_wmma_f32_16x16x4_f32(
        false, a, false, bm, (short)0, acc, false, false);

    const int t = t0 + mr;
    const float4 tcv = *(const float4*)(tc + (size_t)(b * Tn + t) * 4);
    const float tn = tcv.x * tcv.x + tcv.y * tcv.y + tcv.z * tcv.z + tcv.w * tcv.w;

#pragma unroll
    for (int r = 0; r < 8; ++r) {
      const int q = q0 + r + (half << 3);
      float dist2 = fmaxf(pn[r] + tn - 2.0f * d[r], 0.0f);
      const float cost = sqrtf(dist2) - gp[(size_t)(b * Qn + q) * Tn + t];
      Cout[(size_t)(b * Qn + q) * Tn + t] = cost;
      CT[(size_t)(b * Tn + t) * Qn + q] = cost;
    }
  }
}

// ---------------------------------------------------------------------------
// Kernel C: Jonker-Volgenant shortest-augmenting-path assignment.
// One workgroup (256 threads) per batch; n=T=128 rows, m=Q=1024 cols.
// Per-column state in LDS; argmin via LDS tree reduction. The first augmenting
// iteration of each row step always scans row i, so that row is async-copied
// to LDS (global_load_async_to_lds_b128) overlapped with the per-row init.
// ---------------------------------------------------------------------------
__global__ void __launch_bounds__(256) lsa_kernel(
    const float* __restrict__ CT,       // [B,T,Q] : a[i0-1][j-1]
    float*       __restrict__ pred_out, // [B,T]
    float*       __restrict__ tgt_out)  // [B,T]
{
  const int b = blockIdx.x;
  const int tid = threadIdx.x;
  const float* a = CT + (size_t)b * Tn * Qn;

  __shared__ float v[Qn + 1];
  __shared__ float minv[Qn + 1];
  __shared__ float u[Tn + 1];
  __shared__ int   p[Qn + 1];
  __shared__ int   way[Qn + 1];
  __shared__ unsigned char used[Qn + 1];
  __shared__ float rbest[256];
  __shared__ int   rbj[256];
  __shared__ int   col4row[Tn];
  __shared__ float ldsrow[Qn];

  for (int j = tid; j <= Qn; j += 256) { v[j] = 0.0f; p[j] = 0; }
  for (int i = tid; i <= Tn; i += 256) u[i] = 0.0f;
  __syncthreads();

  for (int i = 0; i < Tn; ++i) {
    // Async-copy row i (4KB, 16B per lane) into LDS; overlapped with init.
    {
      const float* gsrc = a + (size_t)i * Qn + (tid << 2);
      unsigned int ldst =
          (unsigned int)(uintptr_t)(&ldsrow[tid << 2]);  // low 32 bits used as LDS address
      asm volatile("global_load_async_to_lds_b128 %0, %1, off"
                   :: "v"(ldst), "v"(gsrc)
                   : "memory");
    }
    if (tid == 0) p[0] = i + 1;
    for (int j = tid; j <= Qn; j += 256) { minv[j] = INFF; used[j] = 0; way[j] = 0; }
    asm volatile("s_wait_asynccnt 0" ::: "memory");
    __syncthreads();

    int j0 = 0;
    bool first = true;
    while (true) {
      if (tid == 0) used[j0] = 1;
      __syncthreads();
      const int i0 = p[j0];
      const float ui0 = u[i0];
      const float* arow = a + (size_t)(i0 - 1) * Qn;

      float best = INFF;
      int bestj = 0;
#pragma unroll
      for (int k = 0; k < 4; ++k) {
        const int j = 1 + tid + (k << 8);  // covers 1..1024
        if (!used[j]) {
          const float aval = first ? ldsrow[j - 1] : arow[j - 1];
          const float cur = aval - ui0 - v[j];
          float mv = minv[j];
          if (cur < mv) { mv = cur; minv[j] = cur; way[j] = j0; }
          if (mv < best) { best = mv; bestj = j; }
        }
      }
      rbest[tid] = best;
      rbj[tid] = bestj;
      __syncthreads();
      for (int s = 128; s > 0; s >>= 1) {
        if (tid < s && rbest[tid + s] < rbest[tid]) {
          rbest[tid] = rbest[tid + s];
          rbj[tid] = rbj[tid + s];
        }
        __syncthreads();
      }
      const float delta = rbest[0];
      const int j1 = rbj[0];

      // u[p[j]] targets are distinct across used j (injective assignment),
      // each j owned by exactly one thread -> race-free plain updates.
      for (int j = tid; j <= Qn; j += 256) {
        if (used[j]) { u[p[j]] += delta; v[j] -= delta; }
        else         { minv[j] -= delta; }
      }
      __syncthreads();
      j0 = j1;
      first = false;
      if (p[j0] == 0) break;
    }

    if (tid == 0) {  // augment along the alternating path
      int jj = j0;
      while (jj != 0) { const int jn = way[jj]; p[jj] = p[jn]; jj = jn; }
    }
    __syncthreads();
  }

  // invert assignment: col4row[row] = query column
  for (int j = 1 + tid; j <= Qn; j += 256)
    if (p[j] > 0) col4row[p[j] - 1] = j - 1;
  __syncthreads();

  // argsort of 128 distinct ints by rank counting
  if (tid < Tn) {
    const int c = col4row[tid];
    int rank = 0;
#pragma unroll 4
    for (int r2 = 0; r2 < Tn; ++r2) rank += (col4row[r2] < c) ? 1 : 0;
    pred_out[b * Tn + rank] = (float)c;
    tgt_out[b * Tn + rank] = (float)tid;
  }
}

// ---------------------------------------------------------------------------
extern "C" void kernel_launch(void* const* d_in, const int* in_sizes, int n_in,
                              void* d_out, int out_size, void* d_ws, size_t ws_size,
                              hipStream_t stream) {
  const float* logits  = (const float*)d_in[0];  // [B,Q,C]
  const float* pcoords = (const float*)d_in[1];  // [B,Q,4]
  const int*   labels  = (const int*)d_in[2];    // [B,T]
  const float* tcoords = (const float*)d_in[3];  // [B,T,4]

  float* Cout     = (float*)d_out;                       // [B,Q,T]
  float* pred_out = Cout + (size_t)Bn * Qn * Tn;         // [B,T]
  float* tgt_out  = pred_out + Bn * Tn;                  // [B,T]

  float* gp = (float*)d_ws;                              // [B,Q,T] probs
  float* CT = gp + (size_t)Bn * Qn * Tn;                 // [B,T,Q] cost^T

  gather_prob_kernel<<<(Bn * Qn) / 8, 256, 0, stream>>>(logits, labels, gp);
  cost_wmma_kernel<<<(Bn * 64) / 8, 256, 0, stream>>>(pcoords, tcoords, gp,
                                                      Cout, CT);
  lsa_kernel<<<Bn, 256, 0, stream>>>(CT, pred_out, tgt_out);
}